// MolecularGCN_56427280335071
// MI455X (gfx1250) — compile-verified
//
#include <hip/hip_runtime.h>
#include <hip/hip_bf16.h>

typedef float v2f __attribute__((ext_vector_type(2)));
typedef float v8f __attribute__((ext_vector_type(8)));

#define DIM 128
#define IN_FEATS 74
#define K_PAD 80   // IN_FEATS padded to multiple of 4 with 8-byte-aligned row stride

// ---------------- utility: zero a float buffer (float4-vectorized) ----------------
__global__ void zero_f4(float4* __restrict__ p, int n4) {
  int i = blockIdx.x * blockDim.x + threadIdx.x;
  if (i < n4) p[i] = make_float4(0.f, 0.f, 0.f, 0.f);
}

// ---------------- pad x [n,74] -> xp [n,80] (cols 74..79 zeroed) ----------------
__global__ void pad_x_kernel(const float* __restrict__ x, float* __restrict__ xp, int n) {
  int i = blockIdx.x * blockDim.x + threadIdx.x;
  if (i < n * K_PAD) {
    int row = i / K_PAD, col = i - row * K_PAD;
    xp[i] = (col < IN_FEATS) ? x[row * IN_FEATS + col] : 0.f;
  }
}

// ---------------- degree accumulation (f32 atomics into L2) ----------------
__global__ void degree_kernel(const int* __restrict__ src, const int* __restrict__ dst,
                              float* __restrict__ deg_out, float* __restrict__ deg_in, int e) {
  int i = blockIdx.x * blockDim.x + threadIdx.x;
  if (i < e) {
    atomicAdd(&deg_out[src[i]], 1.0f);
    atomicAdd(&deg_in[dst[i]], 1.0f);
  }
}

// nrm = rsqrt(max(deg, 1)) in place
__global__ void nrm_kernel(float* __restrict__ d0, float* __restrict__ d1, int n) {
  int i = blockIdx.x * blockDim.x + threadIdx.x;
  if (i < n) {
    d0[i] = rsqrtf(fmaxf(d0[i], 1.0f));
    d1[i] = rsqrtf(fmaxf(d1[i], 1.0f));
  }
}

// ---------------- edge scatter: agg[dst[e], :] += hs[src[e], :] ----------------
// One wave32 per edge; lane l moves dims [4l, 4l+4) as a float4 -> 4 f32 atomics.
// hs already carries the nrm_src scaling (fused into the producing GEMM epilogue).
// Working set (hs 64MB + agg 64MB + edges) is resident in the 192MB L2.
__global__ __launch_bounds__(256) void scatter_kernel(
    const int* __restrict__ src, const int* __restrict__ dst,
    const float* __restrict__ hs, float* __restrict__ agg, int e) {
  int wid  = (int)((blockIdx.x * blockDim.x + threadIdx.x) >> 5);
  int lane = threadIdx.x & 31;
  if (wid >= e) return;
  int s = src[wid];
  int d = dst[wid];
  float4 v = *((const float4*)(hs + (size_t)s * DIM) + lane);
  float* o = agg + (size_t)d * DIM + lane * 4;
  atomicAdd(o + 0, v.x);
  atomicAdd(o + 1, v.y);
  atomicAdd(o + 2, v.z);
  atomicAdd(o + 3, v.w);
}

// ---------------- fused WMMA fp32 GEMM ----------------
// out[m, 0:128] = RELU?( ASCALE?ascale[m]:1 * A[m, 0:K] @ W[0:K, 0:128]
//                        + (BIAS?bias:0) ) * (OSCALE?oscale[m]:1)
// K and all epilogue options are compile-time: no guards, no branches anywhere.
// 256 threads = 8 wave32; each wave owns MT=2 row tiles (32 rows x 128 cols,
// 2x8 v8f accumulators) so every B fragment from LDS feeds 2 WMMAs.
// W (K x 128 <= 64 KB) staged in LDS out of the 320 KB/WGP budget.
template <int K, int MT, bool BIAS, bool ASCALE, bool OSCALE, bool RELU>
__global__ __launch_bounds__(256) void gcn_gemm_wmma(
    const float* __restrict__ A,       // [n][K], row stride K (8B-aligned pairs)
    const float* __restrict__ W,       // [K][128] row-major
    const float* __restrict__ bias,    // [128]
    const float* __restrict__ ascale,  // per-row A scale
    const float* __restrict__ oscale,  // per-row output scale
    float* __restrict__ out) {
  __shared__ float sW[K * DIM];

  int tid = threadIdx.x;
  {
    const float4* W4 = (const float4*)W;
    float4* sW4 = (float4*)sW;
#pragma unroll 4
    for (int i = tid; i < K * DIM / 4; i += 256) sW4[i] = W4[i];
  }
  __syncthreads();

  int wave = tid >> 5;
  int lane = tid & 31;
  int half = lane >> 4;  // 0: K pair {0,1}, 1: K pair {2,3} of each 4-step
  int ln   = lane & 15;
  int m0   = (blockIdx.x * 8 + wave) * (16 * MT);

  // A fragment rows (ISA 7.12.2: 32-bit A 16x4, M = lane&15)
  const float* Arow[MT];
  float as[MT];
#pragma unroll
  for (int t = 0; t < MT; ++t) {
    int m = m0 + t * 16 + ln;
    Arow[t] = A + (size_t)m * K;
    as[t] = ASCALE ? ascale[m] : 1.0f;
  }

  v8f zero = {0.f, 0.f, 0.f, 0.f, 0.f, 0.f, 0.f, 0.f};
  v8f acc[MT][8];
#pragma unroll
  for (int t = 0; t < MT; ++t)
#pragma unroll
    for (int nt = 0; nt < 8; ++nt) acc[t][nt] = zero;

#pragma unroll 2
  for (int kk = 0; kk < K; kk += 4) {
    int kb = kk + half * 2;
    v2f a[MT];
#pragma unroll
    for (int t = 0; t < MT; ++t) {
      v2f av = *(const v2f*)(Arow[t] + kb);  // one global_load_b64
      a[t].x = av.x * as[t];
      a[t].y = av.y * as[t];
    }
#pragma unroll
    for (int nt = 0; nt < 8; ++nt) {
      int n = nt * 16 + ln;  // B 4x16: N = lane&15; lanes 0-15 K{0,1}, 16-31 K{2,3}
      v2f b;
      b.x = sW[(kb + 0) * DIM + n];
      b.y = sW[(kb + 1) * DIM + n];
#pragma unroll
      for (int t = 0; t < MT; ++t) {
        acc[t][nt] = __builtin_amdgcn_wmma_f32_16x16x4_f32(
            /*neg_a=*/false, a[t], /*neg_b=*/false, b,
            /*c_mod=*/(short)0, acc[t][nt], /*reuse_a=*/false, /*reuse_b=*/false);
      }
    }
  }

  // C/D layout: VGPR j holds M = tile_base + j + 8*half, N = nt*16 + (lane&15)
  float bn[8];
#pragma unroll
  for (int nt = 0; nt < 8; ++nt) bn[nt] = BIAS ? bias[nt * 16 + ln] : 0.f;

#pragma unroll
  for (int t = 0; t < MT; ++t) {
    float os[8];
#pragma unroll
    for (int j = 0; j < 8; ++j) {
      int mr = m0 + t * 16 + half * 8 + j;
      os[j] = OSCALE ? oscale[mr] : 1.0f;
    }
#pragma unroll
    for (int nt = 0; nt < 8; ++nt) {
      int n = nt * 16 + ln;
#pragma unroll
      for (int j = 0; j < 8; ++j) {
        int mr = m0 + t * 16 + half * 8 + j;
        float v = acc[t][nt][j] + bn[nt];
        if (RELU) v = fmaxf(v, 0.f);
        out[(size_t)mr * DIM + n] = v * os[j];
      }
    }
  }
}

// ---------------- driver ----------------
extern "C" void kernel_launch(void* const* d_in, const int* in_sizes, int n_in,
                              void* d_out, int out_size, void* d_ws, size_t ws_size,
                              hipStream_t stream) {
  const float* x     = (const float*)d_in[0];
  const int*   src   = (const int*)d_in[1];
  const int*   dst   = (const int*)d_in[2];
  const float* Winit = (const float*)d_in[3];
  const float* W[3]  = {(const float*)d_in[4], (const float*)d_in[6], (const float*)d_in[8]};
  const float* b[3]  = {(const float*)d_in[5], (const float*)d_in[7], (const float*)d_in[9]};
  float* out = (float*)d_out;

  const int n_nodes = in_sizes[0] / IN_FEATS;  // 131072
  const int e       = in_sizes[1];             // 1048576

  // workspace: hs[n*128] | agg[n*128] (reused as padded x) | nsrc[n] | ndst[n] | Wp[80*128]
  float* hs   = (float*)d_ws;
  float* agg  = hs + (size_t)n_nodes * DIM;
  float* nsrc = agg + (size_t)n_nodes * DIM;
  float* ndst = nsrc + n_nodes;
  float* Wp   = ndst + n_nodes;  // 40 KB zero-padded W_init
  float* xp   = agg;             // [n][80]; agg is not live until after the init GEMM

  // degrees -> norms (nsrc/ndst contiguous: single zero pass)
  {
    int n4 = (2 * n_nodes) / 4;
    zero_f4<<<(n4 + 255) / 256, 256, 0, stream>>>((float4*)nsrc, n4);
  }
  degree_kernel<<<(e + 255) / 256, 256, 0, stream>>>(src, dst, nsrc, ndst, e);
  nrm_kernel<<<(n_nodes + 255) / 256, 256, 0, stream>>>(nsrc, ndst, n_nodes);

  // pad x to [n,80] and W_init to [80][128], then
  // hs = (x @ W_init) * nrm_src  (pre-scale fused for the first gather)
  pad_x_kernel<<<(n_nodes * K_PAD + 255) / 256, 256, 0, stream>>>(x, xp, n_nodes);
  {
    int nw4 = K_PAD * DIM / 4;
    zero_f4<<<(nw4 + 255) / 256, 256, 0, stream>>>((float4*)Wp, nw4);
    hipMemcpyAsync(Wp, Winit, sizeof(float) * IN_FEATS * DIM,
                   hipMemcpyDeviceToDevice, stream);
    gcn_gemm_wmma<K_PAD, 2, false, false, true, false>
        <<<n_nodes / 256, 256, 0, stream>>>(xp, Wp, nullptr, nullptr, nsrc, hs);
  }

  const int n4 = (n_nodes * DIM) / 4;
  for (int l = 0; l < 3; ++l) {
    zero_f4<<<(n4 + 255) / 256, 256, 0, stream>>>((float4*)agg, n4);
    scatter_kernel<<<e / 8, 256, 0, stream>>>(src, dst, hs, agg, e);
    // h = relu((agg * nrm_dst) @ W_l + b_l); layers 0,1 additionally pre-scale
    // by nrm_src for the next gather; layer 2 writes the final output unscaled.
    if (l < 2) {
      gcn_gemm_wmma<DIM, 2, true, true, true, true>
          <<<n_nodes / 256, 256, 0, stream>>>(agg, W[l], b[l], ndst, nsrc, hs);
    } else {
      gcn_gemm_wmma<DIM, 2, true, true, false, true>
          <<<n_nodes / 256, 256, 0, stream>>>(agg, W[l], b[l], ndst, nullptr, out);
    }
  }
}